// System_8246337209001
// MI455X (gfx1250) — compile-verified
//
#include <hip/hip_runtime.h>
#include <math.h>

// ---------------- problem constants ----------------
#define D_DIM   131      // 2*PILOT_LEN+3
#define HID     1024
#define UNUM    8        // USER_NUM
#define ROWS    9        // USER_NUM+1
#define NIRS    64
#define OUTC    128      // 2*N_IRS
#define NLAYERS 2
#define SQRT_THR 0.1f    // sqrt(10^(10/10)/1000) = sqrt(0.01)

typedef __attribute__((ext_vector_type(2))) float v2f;
typedef __attribute__((ext_vector_type(8))) float v8f;

// ======================================================================
// Generic fused GEMM + bias + ReLU using V_WMMA_F32_16X16X4_F32.
// Computes O[u] = act( X[u] (MxK) * W[u] (KxN) + b[u] ), M <= 16.
// One wave owns one 16-wide column tile; blockIdx.y = user u.
// A layout (16x4 f32): lanes 0-15 -> row=lane, v0=K0,v1=K1;
//                      lanes16-31 -> row=lane-16, v0=K2,v1=K3.
// B layout (4x16 f32): symmetric, lane%16 = N column.
// C/D (16x16 f32): VGPR e -> rows e (lanes 0-15) and e+8 (lanes 16-31).
// ======================================================================
__global__ __launch_bounds__(128)
void gemm_wmma(const float* __restrict__ X, int ldx, long xs,
               const float* __restrict__ W, const float* __restrict__ Bv,
               long wst, long bst,
               float* __restrict__ O, int ldo, long ost,
               int M, int K, int N, int relu)
{
    const int lane = threadIdx.x & 31;
    const int wave = threadIdx.x >> 5;
    const int u    = blockIdx.y;

    X  += (long)u * xs;
    W  += (long)u * wst;
    Bv += (long)u * bst;
    O  += (long)u * ost;

    const int ncol0 = (blockIdx.x * (blockDim.x >> 5) + wave) * 16;
    if (ncol0 >= N) return;                 // wave-uniform: EXEC stays all-1s

    const int r    = lane & 15;
    const int hi   = lane >> 4;
    const int kb   = hi * 2;
    const int arow = (r < M) ? r : 0;                       // pad rows read row 0
    const int bcol = (ncol0 + r < N) ? (ncol0 + r) : (N-1); // pad cols clamped

    const float* Xp = X + (long)arow * ldx;
    const float* Wp = W + bcol;

    v8f acc = {};
    int kk = 0;
    // main: 16 K-steps per chunk (4 WMMAs), prefetch W row kk+64 (4 chunks ahead)
    for (; kk + 16 <= K; kk += 16) {
        int pk = (kk + 64 < K) ? (kk + 64) : (K - 1);
        __builtin_prefetch(Wp + (long)pk * N, 0, 0);   // -> global_prefetch_b8
#pragma unroll
        for (int s = 0; s < 16; s += 4) {
            const int k = kk + s;
            v2f a, b;
            a.x = Xp[k + kb];
            a.y = Xp[k + kb + 1];
            b.x = Wp[(long)(k + kb)     * N];
            b.y = Wp[(long)(k + kb + 1) * N];
            acc = __builtin_amdgcn_wmma_f32_16x16x4_f32(
                      false, a, false, b, (short)0, acc, false, false);
        }
    }
    // guarded tail (K = 131 / 262 / 393 cases)
    for (; kk < K; kk += 4) {
        v2f a, b;
        const int k0 = kk + kb, k1 = kk + kb + 1;
        a.x = (k0 < K) ? Xp[k0] : 0.0f;
        a.y = (k1 < K) ? Xp[k1] : 0.0f;
        b.x = (k0 < K) ? Wp[(long)k0 * N] : 0.0f;
        b.y = (k1 < K) ? Wp[(long)k1 * N] : 0.0f;
        acc = __builtin_amdgcn_wmma_f32_16x16x4_f32(
                  false, a, false, b, (short)0, acc, false, false);
    }

    const int col = ncol0 + r;
    if (col < N) {
        const float bias = Bv[col];
#pragma unroll
        for (int e = 0; e < 8; ++e) {
            const int m = e + 8 * hi;
            if (m < M) {
                float v = acc[e] + bias;
                if (relu) v = fmaxf(v, 0.0f);
                O[(long)m * ldo + col] = v;
            }
        }
    }
}

// ---------------- small helper kernels ----------------
__global__ void mean_rows(const float* __restrict__ X, float* __restrict__ O,
                          int R, int Dd)
{
    int d = blockIdx.x * blockDim.x + threadIdx.x;
    if (d < Dd) {
        float s = 0.0f;
        for (int r = 0; r < R; ++r) s += X[r * Dd + d];
        O[d] = s / (float)R;
    }
}

// agg[u,d] = max over rows r != u+1 of H[u,r,d];  H is (U, R, Dd)
__global__ void max_excl(const float* __restrict__ H, float* __restrict__ A,
                         int U, int R, int Dd)
{
    int d = blockIdx.x * blockDim.x + threadIdx.x;
    int u = blockIdx.y;
    if (d < Dd) {
        float m = -INFINITY;
        for (int r = 0; r < R; ++r)
            if (r != u + 1)
                m = fmaxf(m, H[((long)u * R + r) * Dd + d]);
        A[u * Dd + d] = m;
    }
}

__global__ void concat2(const float* __restrict__ a, const float* __restrict__ b,
                        float* __restrict__ o, int n)
{
    int i = blockIdx.x * blockDim.x + threadIdx.x;
    if (i < n)            o[i] = a[i];
    else if (i < 2 * n)   o[i] = b[i - n];
}

// mid[u] = [ irs2 (131) | Y[u+1] (131) | agg[u] (131) ]  -> (8, 393)
__global__ void build_mid(const float* __restrict__ irs2, const float* __restrict__ Y,
                          const float* __restrict__ agg, float* __restrict__ mid)
{
    int i = blockIdx.x * blockDim.x + threadIdx.x;
    int u = blockIdx.y;
    if (i < D_DIM)            mid[u * 3 * D_DIM + i] = irs2[i];
    else if (i < 2 * D_DIM)   mid[u * 3 * D_DIM + i] = Y[(u + 1) * D_DIM + (i - D_DIM)];
    else if (i < 3 * D_DIM)   mid[u * 3 * D_DIM + i] = agg[u * D_DIM + (i - 2 * D_DIM)];
}

// final linears + normalizations; out is (1+ROWS) x 128
__global__ __launch_bounds__(OUTC)
void finalize(const float* __restrict__ avg, const float* __restrict__ Y,
              const float* __restrict__ W0, const float* __restrict__ b0,
              const float* __restrict__ W1, const float* __restrict__ b1,
              float* __restrict__ out)
{
    __shared__ float Wm[ROWS][OUTC];
    __shared__ float v0s[OUTC];
    __shared__ float norms[ROWS];
    const int j = threadIdx.x;

    float s = b0[j];
    for (int k = 0; k < D_DIM; ++k) s += avg[k] * W0[k * OUTC + j];
    v0s[j] = s;

    for (int r = 0; r < ROWS; ++r) {
        float t = b1[j];
        for (int k = 0; k < D_DIM; ++k) t += Y[r * D_DIM + k] * W1[k * OUTC + j];
        Wm[r][j] = t;
    }
    __syncthreads();

    if (j < ROWS) {
        float n2 = 0.0f;
        for (int c = 0; c < OUTC; ++c) n2 += Wm[j][c] * Wm[j][c];
        norms[j] = sqrtf(n2);
    }
    __syncthreads();

    if (j < NIRS) {
        float a = v0s[j], b = v0s[j + NIRS];
        float nr = sqrtf(a * a + b * b);
        out[j]        = a / nr;
        out[j + NIRS] = b / nr;
    }
    for (int r = 0; r < ROWS; ++r)
        out[(r + 1) * OUTC + j] = Wm[r][j] / norms[r] * SQRT_THR;
}

// ---------------- host-side sequencing ----------------
static void launch_gemm(const float* X, int ldx, long xs,
                        const float* W, const float* Bv, long wst, long bst,
                        float* O, int ldo, long ost,
                        int M, int K, int N, int relu, int users, hipStream_t st)
{
    const int wavesPerBlock = 4;
    const int ntiles = (N + 15) / 16;
    dim3 grid((ntiles + wavesPerBlock - 1) / wavesPerBlock, users);
    gemm_wmma<<<grid, dim3(128), 0, st>>>(X, ldx, xs, W, Bv, wst, bst,
                                          O, ldo, ost, M, K, N, relu);
}

// Wb = {W0,b0,W1,b1,W2,b2,W3,b3}; dims din->1024->1024->1024->dout, ReLU everywhere
static void run_dnn(const float* const Wb[8],
                    const float* X, int ldx, long xs,
                    float* O, int ldo, long ost,
                    int M, int din, int dout,
                    float* ping, float* pong, long actStride,
                    int users, int batched, hipStream_t st)
{
    const long wst0 = batched ? (long)din * HID : 0;
    const long wstH = batched ? (long)HID * HID : 0;
    const long wstL = batched ? (long)HID * dout : 0;
    const long bstH = batched ? (long)HID : 0;
    const long bstL = batched ? (long)dout : 0;

    launch_gemm(X,    ldx, xs,        Wb[0], Wb[1], wst0, bstH, ping, HID, actStride, M, din, HID, 1, users, st);
    launch_gemm(ping, HID, actStride, Wb[2], Wb[3], wstH, bstH, pong, HID, actStride, M, HID, HID, 1, users, st);
    launch_gemm(pong, HID, actStride, Wb[4], Wb[5], wstH, bstH, ping, HID, actStride, M, HID, HID, 1, users, st);
    launch_gemm(ping, HID, actStride, Wb[6], Wb[7], wstL, bstL, O,    ldo, ost,       M, HID, dout, 1, users, st);
}

extern "C" void kernel_launch(void* const* d_in, const int* in_sizes, int n_in,
                              void* d_out, int out_size, void* d_ws, size_t ws_size,
                              hipStream_t stream)
{
    // JAX pytree flatten order (dict keys sorted at each level):
    //  0: Y
    //  1..8:   in0  (W0,b0,W1,b1,W2,b2,W3,b3)
    //  9..16:  in1
    // 17..24:  irs_dnn
    // per layer l (base = 25 + 32*l):
    //   base+0..7  : agg0_is_irs
    //   base+8..15 : agg0_nn
    //   base+16..23: agg_nn      (lead=8)
    //   base+24..31: agg_not_irs (lead=8)
    // 89: lin0_W  90: lin0_b  91: lin1_W  92: lin1_b
    const float* Yin = (const float*)d_in[0];
    const float* in0[8], *in1[8], *irs[8];
    for (int j = 0; j < 8; ++j) { in0[j] = (const float*)d_in[1 + j];
                                  in1[j] = (const float*)d_in[9 + j];
                                  irs[j] = (const float*)d_in[17 + j]; }
    const float* lin0W = (const float*)d_in[89];
    const float* lin0b = (const float*)d_in[90];
    const float* lin1W = (const float*)d_in[91];
    const float* lin1b = (const float*)d_in[92];

    // workspace layout (floats)
    float* ws     = (float*)d_ws;
    float* Ybuf   = ws;             // 9*131   (pad 1280)
    float* meanb  = ws + 1280;      // 131
    float* avgbuf = ws + 1536;      // 131
    float* irs1   = ws + 1792;      // 131
    float* irs2   = ws + 2048;      // 131
    float* agg0T  = ws + 2304;      // 9*131   (pad 1280)
    float* agg0   = ws + 3584;      // 131
    float* cat0   = ws + 3840;      // 262
    float* aggb   = ws + 4352;      // 8*131
    float* mid    = ws + 5632;      // 8*393
    float* Hbuf   = ws + 8832;      // 8*9*131
    float* pingA  = ws + 18304;     // 8*9*1024
    float* pingB  = ws + 92032;     // 8*9*1024 (end = 165760 floats)

    // avg = mean(Y); avg = dnn(in0, avg); Y = dnn(in1, Y)
    mean_rows<<<dim3(1), dim3(256), 0, stream>>>(Yin, meanb, ROWS, D_DIM);
    run_dnn(in0, meanb, D_DIM, 0, avgbuf, D_DIM, 0, 1, D_DIM, D_DIM,
            pingA, pingB, 0, 1, 0, stream);
    run_dnn(in1, Yin, D_DIM, 0, Ybuf, D_DIM, 0, ROWS, D_DIM, D_DIM,
            pingA, pingB, 0, 1, 0, stream);

    for (int l = 0; l < NLAYERS; ++l) {
        const int base = 25 + 32 * l;
        const float *a0ii[8], *a0nn[8], *annb[8], *anib[8];
        for (int j = 0; j < 8; ++j) { a0ii[j] = (const float*)d_in[base + j];
                                      a0nn[j] = (const float*)d_in[base + 8 + j];
                                      annb[j] = (const float*)d_in[base + 16 + j];
                                      anib[j] = (const float*)d_in[base + 24 + j]; }

        // agg0 = mean(dnn(agg0_nn, Y))
        run_dnn(a0nn, Ybuf, D_DIM, 0, agg0T, D_DIM, 0, ROWS, D_DIM, D_DIM,
                pingA, pingB, 0, 1, 0, stream);
        mean_rows<<<dim3(1), dim3(256), 0, stream>>>(agg0T, agg0, ROWS, D_DIM);

        // avg = dnn(agg0_is_irs, [dnn(irs, avg) | agg0])
        run_dnn(irs, avgbuf, D_DIM, 0, irs1, D_DIM, 0, 1, D_DIM, D_DIM,
                pingA, pingB, 0, 1, 0, stream);
        concat2<<<dim3(2), dim3(256), 0, stream>>>(irs1, agg0, cat0, D_DIM);
        run_dnn(a0ii, cat0, 2 * D_DIM, 0, avgbuf, D_DIM, 0, 1, 2 * D_DIM, D_DIM,
                pingA, pingB, 0, 1, 0, stream);

        // H[u] = dnn(agg_nn[u], Y)  (X shared across users)
        run_dnn(annb, Ybuf, D_DIM, 0, Hbuf, D_DIM, (long)ROWS * D_DIM,
                ROWS, D_DIM, D_DIM, pingA, pingB, (long)ROWS * HID, UNUM, 1, stream);
        max_excl<<<dim3(1, UNUM), dim3(256), 0, stream>>>(Hbuf, aggb, UNUM, ROWS, D_DIM);

        // irs_out = dnn(irs, avg_new); mid = [irs_out | Y[1:] | agg]
        run_dnn(irs, avgbuf, D_DIM, 0, irs2, D_DIM, 0, 1, D_DIM, D_DIM,
                pingA, pingB, 0, 1, 0, stream);
        build_mid<<<dim3(2, UNUM), dim3(256), 0, stream>>>(irs2, Ybuf, aggb, mid);

        // Y[0:8] = dnn(agg_not_irs[u], mid[u])   (row 8 unchanged)
        run_dnn(anib, mid, 3 * D_DIM, 3 * D_DIM, Ybuf, D_DIM, D_DIM,
                1, 3 * D_DIM, D_DIM, pingA, pingB, HID, UNUM, 1, stream);
    }

    finalize<<<dim3(1), dim3(OUTC), 0, stream>>>(avgbuf, Ybuf,
                                                 lin0W, lin0b, lin1W, lin1b,
                                                 (float*)d_out);
}